// DistangledLearn_28535762714739
// MI455X (gfx1250) — compile-verified
//
#include <hip/hip_runtime.h>
#include <cstddef>

// Problem constants (from reference)
#define NN 65536
#define DD 256
#define CC 2048
#define KK 2
#define RR 8
#define BB 64
#define INV_TEMP 20.0f   // 1/0.05
#define TAU_K 0.25f      // TAU/K = 0.5/2
#define EPSF 1e-12f

typedef float v2f __attribute__((ext_vector_type(2)));
typedef float v8f __attribute__((ext_vector_type(8)));

// ---------------- workspace layout (in floats) ----------------
// sums_cr / means_n (in place): [K][C][R][D]   = 8,388,608
// cnt_cr:                        [K][C][R]      = 32,768
// protos_n:                      [C][D]         = 524,288
// outputs:                       [B][C]         = 131,072
// dots:                          [B][K*C*R]     = 2,097,152
#define WS_SUMS   0
#define WS_CNT    (8388608)
#define WS_PROTOS (WS_CNT + 32768)
#define WS_OUT    (WS_PROTOS + 524288)
#define WS_DOTS   (WS_OUT + 131072)
#define WS_TOTAL  (WS_DOTS + 2097152)

// ---------------- kernel 1: zero accumulators + d_out ----------------
__global__ void zero_kernel(float* ws, float* out) {
  size_t total = (size_t)WS_CNT + 32768;  // sums + counts
  for (size_t i = (size_t)blockIdx.x * blockDim.x + threadIdx.x; i < total;
       i += (size_t)gridDim.x * blockDim.x)
    ws[i] = 0.0f;
  if (blockIdx.x == 0 && threadIdx.x == 0) out[0] = 0.0f;
}

// ---------------- kernel 2: scatter ins_memory into sums_cr / cnt_cr ----------------
// one wave (32 lanes) per row; accumulators live in L2 (33.5MB < 192MB)
__global__ void scatter_kernel(const float* __restrict__ ins,
                               const int* __restrict__ labels,
                               const int* __restrict__ irre,
                               float* __restrict__ sums,
                               float* __restrict__ cnt) {
  int wid = (blockIdx.x * blockDim.x + threadIdx.x) >> 5;
  int lane = threadIdx.x & 31;
  int i = wid;  // row index, grid sized exactly to NN waves
  int c = labels[i];
  int r0 = irre[i * 2 + 0];
  int r1 = irre[i * 2 + 1];
  const float4* src = reinterpret_cast<const float4*>(ins + (size_t)i * DD);
  float4 v0 = src[lane];
  float4 v1 = src[lane + 32];
  float* d0 = sums + ((size_t)(0 * CC + c) * RR + r0) * DD;
  float* d1 = sums + ((size_t)(1 * CC + c) * RR + r1) * DD;
  int o = lane * 4;
  atomicAdd(d0 + o + 0, v0.x); atomicAdd(d0 + o + 1, v0.y);
  atomicAdd(d0 + o + 2, v0.z); atomicAdd(d0 + o + 3, v0.w);
  atomicAdd(d0 + 128 + o + 0, v1.x); atomicAdd(d0 + 128 + o + 1, v1.y);
  atomicAdd(d0 + 128 + o + 2, v1.z); atomicAdd(d0 + 128 + o + 3, v1.w);
  atomicAdd(d1 + o + 0, v0.x); atomicAdd(d1 + o + 1, v0.y);
  atomicAdd(d1 + o + 2, v0.z); atomicAdd(d1 + o + 3, v0.w);
  atomicAdd(d1 + 128 + o + 0, v1.x); atomicAdd(d1 + 128 + o + 1, v1.y);
  atomicAdd(d1 + 128 + o + 2, v1.z); atomicAdd(d1 + 128 + o + 3, v1.w);
  if (lane == 0) {
    atomicAdd(cnt + (size_t)c * RR + r0, 1.0f);
    atomicAdd(cnt + (size_t)(CC * RR) + (size_t)c * RR + r1, 1.0f);
  }
}

__device__ __forceinline__ float block_reduce_sum_256(float v, float* sm) {
  int tid = threadIdx.x;
  sm[tid] = v;
  __syncthreads();
  for (int s = 128; s > 0; s >>= 1) {
    if (tid < s) sm[tid] += sm[tid + s];
    __syncthreads();
  }
  float r = sm[0];
  __syncthreads();
  return r;
}

// ---------------- kernel 3: build normalized proto rows for the 64 targets ----------------
// (non-target rows of protos buffer already equal clu_memory via D2D copy; those rows
//  are unit-norm by construction so l2norm is identity on them)
__global__ void protos_kernel(const float* __restrict__ clu,
                              const int* __restrict__ targets,
                              const int* __restrict__ irre_t,
                              const float* __restrict__ sums,
                              const float* __restrict__ cnt,
                              float* __restrict__ protos) {
  __shared__ float sm[256];
  int b = blockIdx.x;
  int d = threadIdx.x;
  int t = targets[b];
  float delta = 0.0f;
#pragma unroll
  for (int k = 0; k < KK; ++k) {
    int rbk = irre_t[b * 2 + k];
    float ps = 0.0f, pc = 0.0f;
#pragma unroll
    for (int r = 0; r < RR; ++r) {
      if (r != rbk) {
        ps += sums[((size_t)(k * CC + t) * RR + r) * DD + d];
        pc += cnt[(size_t)(k * CC + t) * RR + r];
      }
    }
    float m = (pc > 0.0f) ? (ps / fmaxf(pc, 1.0f)) : clu[(size_t)t * DD + d];
    delta += m;
  }
  float p = 0.5f * clu[(size_t)t * DD + d] + TAU_K * delta;
  float ss = block_reduce_sum_256(p * p, sm);
  float scale = 1.0f / fmaxf(sqrtf(ss), EPSF);
  protos[(size_t)t * DD + d] = p * scale;
}

// ---------------- kernel 4: in-place l2-normalize sums_cr rows -> means_n ----------------
__global__ void norm_rows_kernel(float* __restrict__ rows) {
  int wid = (blockIdx.x * blockDim.x + threadIdx.x) >> 5;  // row, exact grid
  int lane = threadIdx.x & 31;
  float4* p = reinterpret_cast<float4*>(rows + (size_t)wid * DD);
  float4 a = p[lane];
  float4 c = p[lane + 32];
  float ss = a.x * a.x + a.y * a.y + a.z * a.z + a.w * a.w +
             c.x * c.x + c.y * c.y + c.z * c.z + c.w * c.w;
#pragma unroll
  for (int off = 16; off > 0; off >>= 1) ss += __shfl_xor(ss, off, 32);
  float s = 1.0f / fmaxf(sqrtf(ss), EPSF);
  a.x *= s; a.y *= s; a.z *= s; a.w *= s;
  c.x *= s; c.y *= s; c.z *= s; c.w *= s;
  p[lane] = a;
  p[lane + 32] = c;
}

// ---------------- kernel 5: out[M=64][Nr] = scale * A[64][256] @ Bm[Nr][256]^T ----------------
// fp32 WMMA 16x16x4, one wave per 16x16 tile, 64 K-steps of 4.
// ISA layout: lane holds (other = lane%16, K = 2*(lane/16)+v) for A/B frags (v2f),
//             C/D: M = g + 8*(lane/16), N = lane%16 (v8f).
__global__ void gemm_bt_kernel(const float* __restrict__ A,
                               const float* __restrict__ Bm,
                               float* __restrict__ out,
                               int nTilesN, float scale) {
  int wid = (blockIdx.x * blockDim.x + threadIdx.x) >> 5;
  int lane = threadIdx.x & 31;
  int mt = wid & 3;          // 64/16 = 4 M-tiles
  int nt = wid >> 2;         // N-tile
  int half = lane >> 4;
  int lm = lane & 15;
  const float* arow = A + (size_t)(mt * 16 + lm) * DD + 2 * half;
  const float* brow = Bm + (size_t)(nt * 16 + lm) * DD + 2 * half;
  v8f acc = {0.f, 0.f, 0.f, 0.f, 0.f, 0.f, 0.f, 0.f};
#pragma unroll 8
  for (int ks = 0; ks < 64; ++ks) {
    v2f av = *reinterpret_cast<const v2f*>(arow + ks * 4);
    v2f bv = *reinterpret_cast<const v2f*>(brow + ks * 4);
    acc = __builtin_amdgcn_wmma_f32_16x16x4_f32(false, av, false, bv,
                                                (short)0, acc, false, false);
  }
  size_t pitch = (size_t)nTilesN * 16;
  size_t col = (size_t)nt * 16 + lm;
#pragma unroll
  for (int g = 0; g < 8; ++g) {
    size_t row = (size_t)(mt * 16 + g + 8 * half);
    out[row * pitch + col] = acc[g] * scale;
  }
}

// ---------------- kernel 6: final loss reduction ----------------
__global__ void finalize_kernel(const float* __restrict__ inputs,
                                const float* __restrict__ clu,
                                const int* __restrict__ targets,
                                const int* __restrict__ irre_t,
                                const float* __restrict__ outs,
                                const float* __restrict__ dots,
                                const float* __restrict__ cnt,
                                float* __restrict__ out) {
  __shared__ float sm[256];
  int b = blockIdx.x;
  int tid = threadIdx.x;
  int t = targets[b];

  // l_sum over exp(outputs[b, :])
  float s = 0.0f;
  for (int c = tid; c < CC; c += 256) s += expf(outs[(size_t)b * CC + c]);
  float lsum = block_reduce_sum_256(s, sm);
  float lpos = expf(outs[(size_t)b * CC + t]);

  float delta_total = 0.0f;
#pragma unroll
  for (int k = 0; k < KK; ++k) {
    int r = irre_t[b * 2 + k];
    float dn = 0.0f, anyf = 0.0f;
    for (int c = tid; c < CC; c += 256) {
      float cv = cnt[(size_t)(k * CC + c) * RR + r];
      if (cv > 0.0f && c != t) {
        dn += expf(dots[(size_t)b * (KK * CC * RR) + (size_t)k * (CC * RR) +
                        (size_t)c * RR + r]);
        anyf += 1.0f;
      }
    }
    dn = block_reduce_sum_256(dn, sm);
    anyf = block_reduce_sum_256(anyf, sm);
    // fallback: exp(dot(inputs[b], l2norm(clu[r])) / TEMP)   (r < 8 < C)
    float q = clu[(size_t)r * DD + tid];
    float ssq = block_reduce_sum_256(q * q, sm);
    float dp = block_reduce_sum_256(inputs[(size_t)b * DD + tid] * q, sm);
    float fb = expf(dp / fmaxf(sqrtf(ssq), EPSF) * INV_TEMP);
    delta_total += (anyf > 0.0f) ? dn : fb;
  }
  lsum += TAU_K * delta_total;
  if (tid == 0) atomicAdd(out, -logf(lpos / lsum) * (1.0f / BB));
}

extern "C" void kernel_launch(void* const* d_in, const int* in_sizes, int n_in,
                              void* d_out, int out_size, void* d_ws, size_t ws_size,
                              hipStream_t stream) {
  const float* inputs     = (const float*)d_in[0];  // (64,256)
  const float* ins_memory = (const float*)d_in[1];  // (65536,256)
  const float* clu_memory = (const float*)d_in[2];  // (2048,256)
  const int*   labels     = (const int*)d_in[3];    // (65536,)
  const int*   irre_lab   = (const int*)d_in[4];    // (65536,2)
  const int*   targets    = (const int*)d_in[5];    // (64,)
  const int*   irre_tgt   = (const int*)d_in[6];    // (64,2)
  (void)d_in; (void)in_sizes; (void)n_in; (void)ws_size;

  float* ws     = (float*)d_ws;
  float* sums   = ws + WS_SUMS;    // becomes means_n in place
  float* cnt    = ws + WS_CNT;
  float* protos = ws + WS_PROTOS;
  float* outs   = ws + WS_OUT;
  float* dots   = ws + WS_DOTS;
  float* outp   = (float*)d_out;

  // 1) zero accumulators + output
  zero_kernel<<<2048, 256, 0, stream>>>(ws, outp);

  // 2) protos buffer starts as clu_memory (non-target rows already unit-norm)
  hipMemcpyAsync(protos, clu_memory, (size_t)CC * DD * sizeof(float),
                 hipMemcpyDeviceToDevice, stream);

  // 3) scatter: one wave per row -> 65536 waves
  scatter_kernel<<<NN / 8, 256, 0, stream>>>(ins_memory, labels, irre_lab, sums, cnt);

  // 4) build 64 normalized proto rows (reads raw sums -> must precede step 5)
  protos_kernel<<<BB, 256, 0, stream>>>(clu_memory, targets, irre_tgt, sums, cnt, protos);

  // 5) in-place normalize sums_cr rows -> means_n (32768 rows, 1 wave each)
  norm_rows_kernel<<<(KK * CC * RR) / 8, 256, 0, stream>>>(sums);

  // 6) outputs = inputs @ protos^T / TEMP : 4 x 128 tiles = 512 waves
  gemm_bt_kernel<<<(4 * (CC / 16)) / 8, 256, 0, stream>>>(inputs, protos, outs,
                                                          CC / 16, INV_TEMP);

  // 7) dots = inputs @ means_n^T / TEMP : B rows = K*C*R = 32768 -> 8192 waves
  gemm_bt_kernel<<<(4 * ((KK * CC * RR) / 16)) / 8, 256, 0, stream>>>(
      inputs, sums, dots, (KK * CC * RR) / 16, INV_TEMP);

  // 8) loss
  finalize_kernel<<<BB, 256, 0, stream>>>(inputs, clu_memory, targets, irre_tgt,
                                          outs, dots, cnt, outp);
}